// FreqAttentionLayerSEFineGrain_20847771254798
// MI455X (gfx1250) — compile-verified
//
#include <hip/hip_runtime.h>
#include <hip/hip_bf16.h>

typedef __attribute__((ext_vector_type(2))) float v2f;
typedef __attribute__((ext_vector_type(4))) float v4f;
typedef __attribute__((ext_vector_type(8))) float v8f;

#define N_   32
#define C_   512
#define HW_  3136          // 56*56
#define F_   8
#define O_   32            // C / R

// ---------------------------------------------------------------------------
// Kernel 1: y[n,f,c] = sum_p x[n,c,p] * dct[f,p]
// fp32 WMMA 16x16x4: A = 16 channels x 4 positions of x, B = 4 positions x
// 16 freqs (8 real + 8 zero via packed mask mul), D = 16x16 accumulator kept
// in VGPRs across the whole K loop. 8 waves split K (392 each), LDS reduce.
// Streams x forward so L2 (192 MB) ends up holding the tail of x for kernel 3.
// ---------------------------------------------------------------------------
__global__ __launch_bounds__(256)
void dct_pool_wmma(const float* __restrict__ x,
                   const float* __restrict__ dct,
                   float* __restrict__ y) {
    __shared__ float red[8 * 256];

    const int n    = blockIdx.y;
    const int c0   = blockIdx.x << 4;          // 16-channel tile
    const int wave = threadIdx.x >> 5;
    const int lane = threadIdx.x & 31;
    const int half = lane >> 4;                // 0: K={0,1}, 1: K={2,3}
    const int m    = lane & 15;                // A: channel row / B: freq col

    // A: lane (m, half) streams x[n, c0+m, k + 2*half .. +1] as float2
    const float* xrow = x + ((size_t)(n * C_ + c0 + m)) * HW_ + 2 * half;
    // B: freq row (m & 7); rows 8..15 zero-padded via mask mul (EXEC stays
    // all-1s around the WMMA, as the ISA requires).
    const float* brow  = dct + (size_t)(m & (F_ - 1)) * HW_ + 2 * half;
    const float  bmask = (m < F_) ? 1.0f : 0.0f;

    const int kbase = wave * (HW_ / 8);        // 392-wide K slice per wave
    v8f acc = {};
    #pragma unroll 4
    for (int k = 0; k < HW_ / 8; k += 4) {
        v2f a = *(const v2f*)(xrow + kbase + k);
        v2f b = *(const v2f*)(brow + kbase + k);
        b.x *= bmask;
        b.y *= bmask;
        acc = __builtin_amdgcn_wmma_f32_16x16x4_f32(
                  /*neg_a=*/false, a, /*neg_b=*/false, b,
                  /*c_mod=*/(short)0, acc, /*reuse_a=*/false, /*reuse_b=*/false);
    }

    // Cross-wave reduction of the 16x16 tile through LDS.
    {
        float* myred = red + wave * 256 + lane;
        #pragma unroll
        for (int r = 0; r < 8; ++r) myred[r * 32] = acc[r];
    }
    __syncthreads();

    float sum = 0.0f;
    #pragma unroll
    for (int w = 0; w < 8; ++w) sum += red[w * 256 + threadIdx.x];

    // Flat tile slot -> (M,N): lanes 0-15: D[r][l]; lanes 16-31: D[r+8][l-16]
    const int r    = threadIdx.x >> 5;
    const int l    = threadIdx.x & 31;
    const int Mrow = r + ((l >> 4) << 3);
    const int Nf   = l & 15;
    if (Nf < F_)
        y[((size_t)n * F_ + Nf) * C_ + c0 + Mrow] = sum;
}

// ---------------------------------------------------------------------------
// Kernel 2 (tiny, one block per image): z = relu(y @ w1), s = sigmoid(z @ w2)
// ---------------------------------------------------------------------------
__global__ __launch_bounds__(256)
void se_mlp(const float* __restrict__ y,
            const float* __restrict__ w1,
            const float* __restrict__ w2,
            float* __restrict__ s) {
    __shared__ float ysh[F_ * C_];   // 16 KB
    __shared__ float zsh[F_ * O_];   // 1 KB

    const int n = blockIdx.x;
    const int t = threadIdx.x;

    for (int i = t; i < F_ * C_; i += 256)
        ysh[i] = y[(size_t)n * (F_ * C_) + i];
    __syncthreads();

    // z[f][o] = relu( sum_c y[n,f,c] * w1[f,o,c] )
    {
        const int f = t >> 5, o = t & 31;
        const float* wrow = w1 + ((size_t)(f * O_ + o)) * C_;
        const float* yrow = ysh + f * C_;
        float acc = 0.0f;
        #pragma unroll 4
        for (int c = 0; c < C_; ++c) acc = fmaf(yrow[c], wrow[c], acc);
        zsh[t] = acc > 0.0f ? acc : 0.0f;
    }
    __syncthreads();

    // s[n,c] = sigmoid( sum_{f,o} z[f,o] * w2[c,f,o] )
    for (int c = t; c < C_; c += 256) {
        const float* wrow = w2 + (size_t)c * (F_ * O_);
        float acc = 0.0f;
        #pragma unroll 4
        for (int i = 0; i < F_ * O_; ++i) acc = fmaf(zsh[i], wrow[i], acc);
        s[(size_t)n * C_ + c] = 1.0f / (1.0f + __expf(-acc));
    }
}

// ---------------------------------------------------------------------------
// Kernel 3: out = x * s[n,c].
// REVERSE-streams x (highest addresses first) so the ~192 MB of x that
// kernel 1 left resident in L2 (LRU tail) is re-read as hits instead of the
// forward-streaming LRU pathology; only the head ~13 MB comes from HBM.
// Output uses NON-TEMPORAL stores so 205 MB of writes don't evict those x
// lines (out is never re-read). float4 per lane; HW % 4 == 0 keeps a float4
// inside one (n,c) row.
// ---------------------------------------------------------------------------
__global__ __launch_bounds__(256)
void scale_out(const float* __restrict__ x,
               const float* __restrict__ s,
               float* __restrict__ out) {
    const unsigned total4 = (unsigned)(N_ * C_ * (HW_ / 4));
    unsigned i4 = blockIdx.x * 256u + threadIdx.x;
    if (i4 >= total4) return;
    unsigned j4 = (total4 - 1u) - i4;          // reverse stream
    unsigned e  = j4 * 4u;
    unsigned nc = e / (unsigned)HW_;           // n*C + c  (32-bit magic div)
    float sv = s[nc];
    v4f v = ((const v4f*)x)[j4];
    v *= sv;
    __builtin_nontemporal_store(v, (v4f*)out + j4);
}

// ---------------------------------------------------------------------------
extern "C" void kernel_launch(void* const* d_in, const int* in_sizes, int n_in,
                              void* d_out, int out_size, void* d_ws, size_t ws_size,
                              hipStream_t stream) {
    const float* x   = (const float*)d_in[0];
    const float* dct = (const float*)d_in[1];
    const float* w1  = (const float*)d_in[2];
    const float* w2  = (const float*)d_in[3];
    float* out = (float*)d_out;

    float* y = (float*)d_ws;             // N*F*C  = 131072 floats (512 KB)
    float* s = y + N_ * F_ * C_;         // N*C    =  16384 floats ( 64 KB)

    dim3 g1(C_ / 16, N_);
    dct_pool_wmma<<<g1, 256, 0, stream>>>(x, dct, y);

    se_mlp<<<N_, 256, 0, stream>>>(y, w1, w2, s);

    const unsigned total4 = (unsigned)(N_ * C_ * (HW_ / 4));
    scale_out<<<(total4 + 255u) / 256u, 256, 0, stream>>>(x, s, out);
}